// h2dLossSep_35502199668826
// MI455X (gfx1250) — compile-verified
//
#include <hip/hip_runtime.h>
#include <cstdint>

// Problem constants (from reference): B=512, 255 layers, sum of counts = 32640.
#define NB     512
#define NL     255
#define NJ     32640
#define TPB    256
#define TILE_F 1024        // floats per tile = TPB * 4 (each lane moves 16B per async op)
#define NTILES 32          // 31 full tiles + 896-float tail (tail = waves 0..6 exactly)
#define DEPTH  4           // async pipeline depth (ring of 4 LDS tiles, 16 KB)

// S(l) = start offset of layer-id l in the flat j ordering; S(l) = l*(511-l)/2.
__device__ __forceinline__ int Sfun(int l) { return (l * (511 - l)) >> 1; }

__global__ __launch_bounds__(TPB)
void hv_loss_partial(const float* __restrict__ d1,
                     const float* __restrict__ hv,
                     const float* __restrict__ mask,
                     float* __restrict__ ws)
{
    __shared__ float sc[256];              // c[0..255] prefix sums
    __shared__ float wl[NL];               // mask^2.5 / (count * B * NL)
    __shared__ float tile[DEPTH][TILE_F];  // 4-deep async stream ring (16 KB)
    __shared__ float red[TPB];

    const int b   = blockIdx.x;
    const int tid = threadIdx.x;
    const float* __restrict__ hrow = hv + (size_t)b * NJ;

    // ---- Phase A: inclusive scan -> c. sc[0]=0, sc[i]=sum_{k<i} d[k]. ----
    sc[tid] = (tid >= 1) ? d1[(size_t)b * NL + (tid - 1)] : 0.0f;
    __syncthreads();
    #pragma unroll
    for (int off = 1; off < 256; off <<= 1) {
        float v = (tid >= off) ? sc[tid - off] : 0.0f;
        __syncthreads();
        sc[tid] += v;
        __syncthreads();
    }

    // ---- Phase B: per-layer weight with all normalization folded in. ----
    if (tid < NL) {
        float mm = mask[(size_t)b * NL + tid];
        // count for layer-id l is (255 - l); total scale = count * B * NL
        float denom = (float)(NL - tid) * (float)NB * (float)NL;
        wl[tid] = powf(mm, 2.5f) / denom;
    }
    __syncthreads();

    // ---- Phase C: 4-deep async global->LDS stream of the hvec row. ----
    // Each lane async-copies exactly the 16B it will later consume, so a
    // per-wave s_wait_asynccnt is sufficient (no barriers in the hot loop).
    const uint32_t lds_base = (uint32_t)(uintptr_t)(&tile[0][tid * 4]);

    auto stage = [&](int buf, int t) {
        int j = t * TILE_F + tid * 4;
        if (j > NJ - 4) j = NJ - 4;        // clamp tail lanes in-bounds (dup data, unused)
        const float* g = hrow + j;
        uint32_t lds = lds_base + (uint32_t)buf * (TILE_F * 4u);
        asm volatile("global_load_async_to_lds_b128 %0, %1, off"
                     :: "v"(lds), "v"(g) : "memory");
    };

    stage(0, 0);
    stage(1, 1);
    stage(2, 2);

    float acc = 0.0f;
    for (int t = 0; t < NTILES; ++t) {
        if (t + (DEPTH - 1) < NTILES) {
            stage((t + (DEPTH - 1)) & (DEPTH - 1), t + (DEPTH - 1));
            asm volatile("s_wait_asynccnt 0x3" ::: "memory");  // tile t landed (in-order)
        } else {
            asm volatile("s_wait_asynccnt 0x0" ::: "memory");  // drain remaining tail tiles
        }

        int j0 = t * TILE_F + tid * 4;
        if (j0 < NJ) {   // wave-uniform: only wave 7 of tile 31 is excluded
            const float* tl = &tile[t & (DEPTH - 1)][tid * 4];
            float h0 = tl[0], h1 = tl[1], h2 = tl[2], h3 = tl[3];

            // Analytic inverse j -> layer id: largest l with S(l) <= j0.
            float fj = (float)j0;
            int l = (int)((511.0f - sqrtf(261121.0f - 8.0f * fj)) * 0.5f);
            if (l < 0)   l = 0;
            if (l > 254) l = 254;
            while (Sfun(l + 1) <= j0) ++l;     // +/-1 ulp fixup
            while (Sfun(l) > j0)      --l;

            int   base  = Sfun(l);
            int   nextS = base + (NL - l);
            float w     = wl[l];
            float hvals[4] = {h0, h1, h2, h3};

            #pragma unroll
            for (int k = 0; k < 4; ++k) {
                int j = j0 + k;
                while (j >= nextS) {           // layer boundary crossings
                    ++l;
                    base  = nextS;
                    nextS = base + (NL - l);
                    w     = wl[l];
                }
                int lo = j - base;             // hi = lo + (l+1)
                float diff = sc[lo + l + 1] - sc[lo] - hvals[k];
                acc = fmaf(diff * w, diff, acc);
            }
        }
    }

    // ---- Phase D: deterministic in-block tree reduction. ----
    red[tid] = acc;
    __syncthreads();
    #pragma unroll
    for (int s = TPB / 2; s > 0; s >>= 1) {
        if (tid < s) red[tid] += red[tid + s];
        __syncthreads();
    }
    if (tid == 0) ws[b] = red[0];
}

__global__ __launch_bounds__(256)
void hv_loss_final(const float* __restrict__ ws, float* __restrict__ out)
{
    __shared__ float red[256];
    int tid = threadIdx.x;
    red[tid] = ws[tid] + ws[tid + 256];        // 512 partials -> 256
    __syncthreads();
    #pragma unroll
    for (int s = 128; s > 0; s >>= 1) {
        if (tid < s) red[tid] += red[tid + s];
        __syncthreads();
    }
    if (tid == 0) out[0] = red[0];
}

extern "C" void kernel_launch(void* const* d_in, const int* in_sizes, int n_in,
                              void* d_out, int out_size, void* d_ws, size_t ws_size,
                              hipStream_t stream)
{
    (void)in_sizes; (void)n_in; (void)out_size; (void)ws_size;
    const float* d1   = (const float*)d_in[0];   // (512,1,255) f32
    const float* hv   = (const float*)d_in[1];   // (512,32640) f32
    const float* mask = (const float*)d_in[2];   // (512,255)   f32
    float* ws  = (float*)d_ws;                   // 512 partials (2 KB)
    float* out = (float*)d_out;                  // scalar f32

    hv_loss_partial<<<NB, TPB, 0, stream>>>(d1, hv, mask, ws);
    hv_loss_final<<<1, 256, 0, stream>>>(ws, out);
}